// GraphAttentionLayer_33311766347940
// MI455X (gfx1250) — compile-verified
//
#include <hip/hip_runtime.h>
#include <math.h>

#define B_SZ 8
#define N_SZ 2048
#define D_SZ 128
#define ALPHA 0.2f

typedef __attribute__((ext_vector_type(2))) float v2f;
typedef __attribute__((ext_vector_type(8))) float v8f;

// ---------------------------------------------------------------------------
// Kernel 1: Wh = h @ W  (fp32 WMMA 16x16x4), plus s_src = Wh@a[:D], s_dst = Wh@a[D:]
// One wave (32 threads) per 16x128 Wh tile.
// A layout (16x4 f32): lane L, vgpr v -> (M = L%16, K = 2*(L/16) + v)
// B layout (4x16 f32): lane L, vgpr v -> (K = 2*(L/16) + v, N = L%16)
// C/D layout (16x16 f32): lane L, vgpr v -> (M = v + 8*(L/16), N = L%16)
// ---------------------------------------------------------------------------
__global__ void k_wh(const float* __restrict__ h, const float* __restrict__ W,
                     const float* __restrict__ a, float* __restrict__ Wh,
                     float* __restrict__ s_src, float* __restrict__ s_dst) {
  __shared__ float lds[16 * D_SZ];

  const int blk  = blockIdx.x;
  const int b    = blk / (N_SZ / 16);
  const int i0   = (blk % (N_SZ / 16)) * 16;
  const int lane = threadIdx.x & 31;
  const int half = lane >> 4;
  const int lm   = lane & 15;

  const size_t bbase = (size_t)b * N_SZ;
  const float* hrow  = h + (bbase + i0 + lm) * D_SZ;

  v8f acc[8] = {};
#pragma unroll 4
  for (int k = 0; k < 32; ++k) {
    const float* ap = hrow + 4 * k + 2 * half;
    v2f A = {ap[0], ap[1]};
    const float* brow = W + (size_t)(4 * k + 2 * half) * D_SZ + lm;
#pragma unroll
    for (int t = 0; t < 8; ++t) {
      v2f Bv = {brow[16 * t], brow[16 * t + D_SZ]};
      acc[t] = __builtin_amdgcn_wmma_f32_16x16x4_f32(false, A, false, Bv,
                                                     (short)0, acc[t], false, false);
    }
  }

  // Store Wh tile to global + LDS (for score reduction).
  float* wout = Wh + (bbase + i0) * D_SZ;
#pragma unroll
  for (int t = 0; t < 8; ++t) {
#pragma unroll
    for (int v = 0; v < 8; ++v) {
      const int row = v + 8 * half;
      const int col = 16 * t + lm;
      wout[(size_t)row * D_SZ + col] = acc[t][v];
      lds[row * D_SZ + col] = acc[t][v];
    }
  }
  __syncthreads();

  // lanes 0..15 -> s_src for row lm ; lanes 16..31 -> s_dst for row lm.
  const float* av = a + half * D_SZ;
  float s = 0.f;
#pragma unroll 8
  for (int d = 0; d < D_SZ; ++d) s += lds[lm * D_SZ + d] * av[d];
  float* so = half ? s_dst : s_src;
  so[bbase + i0 + lm] = s;
}

// ---------------------------------------------------------------------------
// Kernel 2: per-row online softmax stats over masked e = lrelu(s_i + s_j).
// One wave per row; 8 waves per block. Writes m_i and 1/Z_i.
// ---------------------------------------------------------------------------
__global__ void k_stats(const int* __restrict__ adj, const float* __restrict__ s_src,
                        const float* __restrict__ s_dst, float* __restrict__ m_out,
                        float* __restrict__ zinv_out) {
  const int wid  = threadIdx.x >> 5;
  const int lane = threadIdx.x & 31;
  const int row  = blockIdx.x * 8 + wid;  // flat (b*N + i)
  const int b    = row / N_SZ;

  const int* arow  = adj + (size_t)row * N_SZ;
  const float* sdr = s_dst + (size_t)b * N_SZ;
  const float si   = s_src[row];

  float m = -INFINITY, l = 0.f;
  for (int j = lane; j < N_SZ; j += 32) {
    if (arow[j] != 0) {
      float x = si + sdr[j];
      float e = x > 0.f ? x : ALPHA * x;
      if (e > m) {
        l = l * __expf(m - e) + 1.f;  // expf(-inf)=0 handles first hit
        m = e;
      } else {
        l += __expf(e - m);
      }
    }
  }
  // wave32 tree reduction of (m, l)
#pragma unroll
  for (int off = 16; off > 0; off >>= 1) {
    float m2 = __shfl_xor(m, off, 32);
    float l2 = __shfl_xor(l, off, 32);
    float mn = fmaxf(m, m2);
    float t1 = (l > 0.f) ? l * __expf(m - mn) : 0.f;
    float t2 = (l2 > 0.f) ? l2 * __expf(m2 - mn) : 0.f;
    m = mn;
    l = t1 + t2;
  }
  if (lane == 0) {
    m_out[row]    = m;
    zinv_out[row] = (l > 0.f) ? 1.f / l : 0.f;
  }
}

// ---------------------------------------------------------------------------
// Kernel 3: h' = softmax(e) @ Wh via fp32 WMMA, P built on the fly.
// Block = 4 waves = 64 output rows; Wh j-tile (16x128) staged in LDS once and
// shared by all 4 waves. adj read directly (second and last HBM pass).
// ---------------------------------------------------------------------------
__global__ void k_aggr(const int* __restrict__ adj, const float* __restrict__ Wh,
                       const float* __restrict__ s_src, const float* __restrict__ s_dst,
                       const float* __restrict__ m_in, const float* __restrict__ zinv,
                       float* __restrict__ out) {
  __shared__ float whs[16 * D_SZ];  // 8KB staged Wh j-tile
  __shared__ float sds[16];         // staged s_dst for the j-tile

  const int blk  = blockIdx.x;
  const int b    = blk / (N_SZ / 64);
  const int gtl  = blk % (N_SZ / 64);
  const int wid  = threadIdx.x >> 5;
  const int lane = threadIdx.x & 31;
  const int half = lane >> 4;
  const int lm   = lane & 15;

  const size_t bbase = (size_t)b * N_SZ;
  const int i0   = gtl * 64 + wid * 16;  // this wave's 16-row tile
  const int rowA = i0 + lm;              // A-layout row handled by this lane

  const float si = s_src[bbase + rowA];
  const float mi = m_in[bbase + rowA];
  const int* arow = adj + (bbase + rowA) * N_SZ;

  v8f acc[8] = {};

  for (int jt = 0; jt < N_SZ / 16; ++jt) {
    const int jbase = jt * 16;
    __syncthreads();  // protect whs/sds from previous iteration readers
    {
      const float4* src = (const float4*)(Wh + (bbase + jbase) * D_SZ);
      float4* dst = (float4*)whs;
#pragma unroll
      for (int u = 0; u < 4; ++u) dst[threadIdx.x + 128 * u] = src[threadIdx.x + 128 * u];
      if (threadIdx.x < 16) sds[threadIdx.x] = s_dst[bbase + jbase + threadIdx.x];
    }
    __syncthreads();

    // Build P~ = adj * exp(lrelu(si + sj) - mi) directly in WMMA A layout.
    float pv[4][2];
#pragma unroll
    for (int k = 0; k < 4; ++k) {
      const int joff = 4 * k + 2 * half;
      const int a0 = arow[jbase + joff];
      const int a1 = arow[jbase + joff + 1];
      float x0 = si + sds[joff];
      float x1 = si + sds[joff + 1];
      float e0 = x0 > 0.f ? x0 : ALPHA * x0;
      float e1 = x1 > 0.f ? x1 : ALPHA * x1;
      pv[k][0] = a0 ? __expf(e0 - mi) : 0.f;
      pv[k][1] = a1 ? __expf(e1 - mi) : 0.f;
    }

#pragma unroll
    for (int k = 0; k < 4; ++k) {
      v2f A = {pv[k][0], pv[k][1]};
      const float* bp = whs + (4 * k + 2 * half) * D_SZ + lm;
#pragma unroll
      for (int t = 0; t < 8; ++t) {
        v2f Bv = {bp[16 * t], bp[16 * t + D_SZ]};
        acc[t] = __builtin_amdgcn_wmma_f32_16x16x4_f32(false, A, false, Bv,
                                                       (short)0, acc[t], false, false);
      }
    }
  }

  // Scale rows by 1/Z and store (C/D layout: row = v + 8*half, col = 16t + lm).
  float zv[8];
#pragma unroll
  for (int v = 0; v < 8; ++v) zv[v] = zinv[bbase + i0 + v + 8 * half];
  float* orow = out + (bbase + i0) * D_SZ;
#pragma unroll
  for (int t = 0; t < 8; ++t) {
#pragma unroll
    for (int v = 0; v < 8; ++v) {
      orow[(size_t)(v + 8 * half) * D_SZ + 16 * t + lm] = acc[t][v] * zv[v];
    }
  }
}

// ---------------------------------------------------------------------------
extern "C" void kernel_launch(void* const* d_in, const int* in_sizes, int n_in,
                              void* d_out, int out_size, void* d_ws, size_t ws_size,
                              hipStream_t stream) {
  const float* h   = (const float*)d_in[0];
  const int*   adj = (const int*)d_in[1];
  const float* W   = (const float*)d_in[2];
  const float* a   = (const float*)d_in[3];
  float* out = (float*)d_out;

  float* ws    = (float*)d_ws;
  float* Wh    = ws;                                   // B*N*D floats (8 MB)
  float* s_src = Wh + (size_t)B_SZ * N_SZ * D_SZ;      // B*N
  float* s_dst = s_src + (size_t)B_SZ * N_SZ;          // B*N
  float* m     = s_dst + (size_t)B_SZ * N_SZ;          // B*N
  float* zi    = m + (size_t)B_SZ * N_SZ;              // B*N

  k_wh<<<B_SZ * (N_SZ / 16), 32, 0, stream>>>(h, W, a, Wh, s_src, s_dst);
  k_stats<<<(B_SZ * N_SZ) / 8, 256, 0, stream>>>(adj, s_src, s_dst, m, zi);
  k_aggr<<<B_SZ * (N_SZ / 64), 128, 0, stream>>>(adj, Wh, s_src, s_dst, m, zi, out);
}